// RDModel_56779467653666
// MI455X (gfx1250) — compile-verified
//
#include <hip/hip_runtime.h>

typedef __attribute__((ext_vector_type(2))) float v2f;
typedef __attribute__((ext_vector_type(8))) float v8f;

#define RANK     32
#define RANK_EFF 16
#define HORIZON  8
#define D_OUTPUT 8192
#define BATCH    8

// One wave32 workgroup per batch element. Chain of 16x16 f32 matmuls on the
// WMMA pipe: each 16x16x16 product = 4 accumulating V_WMMA_F32_16X16X4_F32.
__global__ __launch_bounds__(32) void mps_chain_kernel(
    const float* __restrict__ x,
    const int*   __restrict__ y,
    const float* __restrict__ G,
    const float* __restrict__ a_full,
    const float* __restrict__ b_full,
    const int*   __restrict__ ids_g,
    const int*   __restrict__ ids_a,
    const int*   __restrict__ ids_b,
    float*       __restrict__ out)
{
    __shared__ float P[16][17];   // running product, padded (no bank conflicts)
    __shared__ float M[16][17];   // current gathered matrix
    __shared__ int   sidg[16];
    __shared__ float sa[16], sb[16];
    __shared__ float red[32];

    const int lane = threadIdx.x;   // 0..31, wave32
    const int b    = blockIdx.x;    // batch element
    const float xb = x[b];          // D_INPUT == 1 -> scalar scale

    if (lane < 16) {
        sidg[lane] = ids_g[lane];
        sa[lane]   = a_full[ids_a[lane]];
        sb[lane]   = b_full[ids_b[lane]];
    }
    __syncthreads();

    // h = 0: gather first matrix straight into the running product P.
    {
        const int o = y[b * HORIZON + 0];
        #pragma unroll
        for (int i = 0; i < 8; ++i) {
            const int idx = lane + 32 * i;          // 256 elements / 32 lanes
            const int p = idx >> 4, q = idx & 15;
            const long long flat =
                (((long long)(sidg[p] * RANK + sidg[q]) * HORIZON + 0) * D_OUTPUT) + o;
            P[p][q] = G[flat] * xb;
        }
    }
    __syncthreads();

    const int hi = lane >> 4;   // 0 for lanes 0-15, 1 for lanes 16-31
    const int ln = lane & 15;

    for (int h = 1; h < HORIZON; ++h) {
        // Gather M_h[p][q] = G[ids_g[p], ids_g[q], h, y[b,h]] * x[b]
        const int o = y[b * HORIZON + h];
        #pragma unroll
        for (int i = 0; i < 8; ++i) {
            const int idx = lane + 32 * i;
            const int p = idx >> 4, q = idx & 15;
            const long long flat =
                (((long long)(sidg[p] * RANK + sidg[q]) * HORIZON + h) * D_OUTPUT) + o;
            M[p][q] = G[flat] * xb;
        }
        __syncthreads();

        // P_new = P * M_h  : 16x16x16 f32 = 4 accumulating K=4 WMMAs.
        v8f c = {};
        #pragma unroll
        for (int kc = 0; kc < 4; ++kc) {
            v2f af, bf;
            // A frag (16x4): m = ln, k = kc*4 + vgpr + 2*hi  (ISA layout)
            af.x = P[ln][kc * 4 + 0 + 2 * hi];
            af.y = P[ln][kc * 4 + 1 + 2 * hi];
            // B frag (4x16): n = ln, k = kc*4 + vgpr + 2*hi  (mirrored layout)
            bf.x = M[kc * 4 + 0 + 2 * hi][ln];
            bf.y = M[kc * 4 + 1 + 2 * hi][ln];
            c = __builtin_amdgcn_wmma_f32_16x16x4_f32(
                    /*neg_a=*/false, af, /*neg_b=*/false, bf,
                    /*c_mod=*/(short)0, c, /*reuse_a=*/false, /*reuse_b=*/false);
        }
        __syncthreads();   // all lanes done reading P before overwrite

        // C/D layout back to LDS: row = r + 8*hi, col = ln
        #pragma unroll
        for (int r = 0; r < 8; ++r)
            P[r + 8 * hi][ln] = c[r];
        __syncthreads();
    }

    // res[b] = sum_{p,q} a[p] * P[p][q] * b[q]
    float partial = 0.0f;
    #pragma unroll
    for (int i = lane; i < 256; i += 32) {
        const int p = i >> 4, q = i & 15;
        partial += sa[p] * P[p][q] * sb[q];
    }
    red[lane] = partial;
    __syncthreads();
    if (lane == 0) {
        float s = 0.0f;
        #pragma unroll
        for (int i = 0; i < 32; ++i) s += red[i];
        out[b] = s;
    }
}

extern "C" void kernel_launch(void* const* d_in, const int* in_sizes, int n_in,
                              void* d_out, int out_size, void* d_ws, size_t ws_size,
                              hipStream_t stream) {
    // setup_inputs() order: x, y, G, a_full, b_full, ids_g, ids_a, ids_b
    const float* x      = (const float*)d_in[0];
    const int*   y      = (const int*)  d_in[1];
    const float* G      = (const float*)d_in[2];
    const float* a_full = (const float*)d_in[3];
    const float* b_full = (const float*)d_in[4];
    const int*   ids_g  = (const int*)  d_in[5];
    const int*   ids_a  = (const int*)  d_in[6];
    const int*   ids_b  = (const int*)  d_in[7];
    float* out = (float*)d_out;

    mps_chain_kernel<<<BATCH, 32, 0, stream>>>(
        x, y, G, a_full, b_full, ids_g, ids_a, ids_b, out);
}